// PGA_Unfold_J10_decay_41772851921184
// MI455X (gfx1250) — compile-verified
//
#include <hip/hip_runtime.h>
#include <math.h>

typedef __attribute__((ext_vector_type(2))) float v2f;
typedef __attribute__((ext_vector_type(8))) float v8f;

#define K_SIGMA2 1.0f
#define K_LN2    0.69314718055994530942f

struct C { float re, im; };
__device__ __forceinline__ C cmk(float r, float i){ C c; c.re=r; c.im=i; return c; }
__device__ __forceinline__ C cadd(C a, C b){ return cmk(a.re+b.re, a.im+b.im); }
__device__ __forceinline__ C csub(C a, C b){ return cmk(a.re-b.re, a.im-b.im); }
__device__ __forceinline__ C cmul(C a, C b){ return cmk(fmaf(a.re,b.re,-(a.im*b.im)), fmaf(a.re,b.im, a.im*b.re)); }
// a * conj(b)
__device__ __forceinline__ C cmulc(C a, C b){ return cmk(fmaf(a.re,b.re, a.im*b.im), fmaf(a.im,b.re,-(a.re*b.im))); }
__device__ __forceinline__ C cscale(C a, float s){ return cmk(a.re*s, a.im*s); }
__device__ __forceinline__ C cdiv(C a, C b){
  float inv = 1.0f/(b.re*b.re + b.im*b.im);
  return cmk((a.re*b.re + a.im*b.im)*inv, (a.im*b.re - a.re*b.im)*inv);
}
__device__ __forceinline__ float wsum32(float v){
  #pragma unroll
  for (int off = 16; off > 0; off >>= 1) v += __shfl_xor(v, off, 32);
  return v;
}

// ---------------- Mm = A^H R A precompute (batch-invariant) ----------------
__global__ void k_mm_RA(const float* __restrict__ Rv, const float* __restrict__ Av,
                        float* __restrict__ Tv) {
  int idx = blockIdx.x * blockDim.x + threadIdx.x;
  if (idx >= 64*64) return;
  int i = idx >> 6, j = idx & 63;
  float tre = 0.f, tim = 0.f;
  for (int k = 0; k < 64; ++k) {
    float ar = Rv[(i*64+k)*2], ai = Rv[(i*64+k)*2+1];
    float br = Av[(k*64+j)*2], bi = Av[(k*64+j)*2+1];
    tre = fmaf(ar, br, fmaf(-ai, bi, tre));
    tim = fmaf(ar, bi, fmaf( ai, br, tim));
  }
  Tv[idx*2] = tre; Tv[idx*2+1] = tim;
}

__global__ void k_mm_AHT(const float* __restrict__ Av, const float* __restrict__ Tv,
                         float* __restrict__ Mmv) {
  int idx = blockIdx.x * blockDim.x + threadIdx.x;
  if (idx >= 64*64) return;
  int i = idx >> 6, j = idx & 63;
  float tre = 0.f, tim = 0.f;
  for (int k = 0; k < 64; ++k) {
    float ar = Av[(k*64+i)*2], ai = Av[(k*64+i)*2+1];   // A[k][i]
    float br = Tv[(k*64+j)*2], bi = Tv[(k*64+j)*2+1];   // T[k][j]
    // conj(a)*b
    tre = fmaf(ar, br, fmaf( ai, bi, tre));
    tim = fmaf(ar, bi, fmaf(-ai, br, tim));
  }
  Mmv[idx*2] = tre; Mmv[idx*2+1] = tim;
}

// F is kept in LDS in fused B-matrix layout: row n (16 floats) = [re(0..7) | im(0..7)]
#define FRE(n,r) sFC[(n)*16 + (r)]
#define FIM(n,r) sFC[(n)*16 + 8 + (r)]

// ---------------- MmF = Mm @ F via fp32 WMMA (complex via 2 real chains) ----------------
// B matrix (64x16) = fused F rows (already in WMMA B layout -> branch-free loads)
// P = Mm_re @ B, Q = Mm_im @ B, stored as [64][16] in LDS.
// MmF_re(n,r) = P[n][r]   - Q[n][r+8]
// MmF_im(n,r) = P[n][r+8] + Q[n][r]
__device__ __forceinline__ void compute_MmF(const float* sMmRe, const float* sMmIm,
                                            const float* sFC,
                                            float* sP, float* sQ, int lane) {
  v8f accP[4], accQ[4];
  #pragma unroll
  for (int t = 0; t < 4; ++t) {
    #pragma unroll
    for (int v = 0; v < 8; ++v) { accP[t][v] = 0.0f; accQ[t][v] = 0.0f; }
  }
  const int col  = lane & 15;
  const int half = lane >> 4;
  for (int kb = 0; kb < 16; ++kb) {
    const int k0 = kb*4 + half*2;
    v2f bf;                                  // unconditional: one ds_load_2addr
    bf.x = sFC[k0*16 + col];
    bf.y = sFC[(k0+1)*16 + col];
    #pragma unroll
    for (int t = 0; t < 4; ++t) {
      const int m = t*16 + col;
      v2f ar; ar.x = sMmRe[m*64 + k0]; ar.y = sMmRe[m*64 + k0 + 1];
      v2f ai; ai.x = sMmIm[m*64 + k0]; ai.y = sMmIm[m*64 + k0 + 1];
      accP[t] = __builtin_amdgcn_wmma_f32_16x16x4_f32(false, ar, false, bf, (short)0, accP[t], false, false);
      accQ[t] = __builtin_amdgcn_wmma_f32_16x16x4_f32(false, ai, false, bf, (short)0, accQ[t], false, false);
    }
  }
  __syncthreads();
  #pragma unroll
  for (int t = 0; t < 4; ++t) {
    #pragma unroll
    for (int v = 0; v < 8; ++v) {
      const int n = t*16 + v + half*8;       // C/D layout: vgpr v -> row v (+8 for hi half)
      sP[n*16 + col] = accP[t][v];
      sQ[n*16 + col] = accQ[t][v];
    }
  }
  __syncthreads();
}

// ---------------- main PGA unfolding: 1 wave32 per batch problem ----------------
__global__ void __launch_bounds__(32)
k_pga(const float* __restrict__ Hg, const float* __restrict__ Mmg,
      const float* __restrict__ ss, const float* __restrict__ F0g,
      const float* __restrict__ W0g, const float* __restrict__ PtG,
      float* __restrict__ outg, int B)
{
  __shared__ float sMmRe[4096], sMmIm[4096];
  __shared__ float sFC[1024];                // F fused layout [64][16] = [re8|im8]
  __shared__ float sHre[256],  sHim[256];
  __shared__ float sWre[32],   sWim[32];
  __shared__ float sP[1024],   sQ[1024];
  __shared__ float sVre[64],   sVim[64];     // V = W W^H
  __shared__ float sMre[64],   sMim[64];     // FHMF = F^H Mm F
  __shared__ float sUre[32],   sUim[32];     // u = H^H F   [4][8]
  __shared__ float sUVre[32],  sUVim[32];
  __shared__ float sUKre[32],  sUKim[32];    // uVmk
  __shared__ float sI1re[4], sI1im[4], sI2re[4], sI2im[4];
  __shared__ float sGre[32],  sGim[32];      // g = F^H h_m  [4][8]
  __shared__ float sGHre[16], sGHim[16];     // gHW [4][4]
  __shared__ float sFcRe[4],  sFcIm[4];      // diag/denom

  const int lane = threadIdx.x;
  const int b = blockIdx.x;
  if (b >= B) return;
  const float Pt = PtG[0];

  for (int e = lane; e < 4096; e += 32) { sMmRe[e] = Mmg[2*e];   sMmIm[e] = Mmg[2*e+1]; }
  { const float* Hb = Hg  + (size_t)b*256*2;
    for (int e = lane; e < 256; e += 32) { sHre[e] = Hb[2*e]; sHim[e] = Hb[2*e+1]; } }
  { const float* Fb = F0g + (size_t)b*512*2;
    for (int e = lane; e < 512; e += 32) {
      int n = e >> 3, r = e & 7;
      FRE(n,r) = Fb[2*e]; FIM(n,r) = Fb[2*e+1];
    } }
  { const float* Wb = W0g + (size_t)b*32*2;
    sWre[lane] = Wb[2*lane]; sWim[lane] = Wb[2*lane+1]; }
  __syncthreads();

  for (int ii = 0; ii < 10; ++ii) {
    // ---------------- inner loop: F updates ----------------
    for (int jj = 0; jj < 5; ++jj) {
      compute_MmF(sMmRe, sMmIm, sFC, sP, sQ, lane);

      // V = W W^H   (8x8)
      for (int e = lane; e < 64; e += 32) {
        int i = e >> 3, j = e & 7;
        C acc = cmk(0.f, 0.f);
        #pragma unroll
        for (int m = 0; m < 4; ++m)
          acc = cadd(acc, cmulc(cmk(sWre[i*4+m], sWim[i*4+m]), cmk(sWre[j*4+m], sWim[j*4+m])));
        sVre[e] = acc.re; sVim[e] = acc.im;
      }
      // u[m][r] = sum_n conj(H[m][n]) F[n][r]
      {
        int m = lane >> 3, r = lane & 7;
        C acc = cmk(0.f, 0.f);
        for (int n = 0; n < 64; ++n)
          acc = cadd(acc, cmulc(cmk(FRE(n,r), FIM(n,r)), cmk(sHre[m*64+n], sHim[m*64+n])));
        sUre[lane] = acc.re; sUim[lane] = acc.im;
      }
      // FHMF[i][j] = sum_n conj(F[n][i]) MmF[n][j]
      for (int e = lane; e < 64; e += 32) {
        int i = e >> 3, j = e & 7;
        C acc = cmk(0.f, 0.f);
        for (int n = 0; n < 64; ++n) {
          C mf = cmk(sP[n*16+j] - sQ[n*16+j+8], sP[n*16+j+8] + sQ[n*16+j]);
          acc = cadd(acc, cmulc(mf, cmk(FRE(n,i), FIM(n,i))));
        }
        sMre[e] = acc.re; sMim[e] = acc.im;
      }
      __syncthreads();

      // tr = sum_{i,j} FHMF[i][j] * V[j][i]   (all-lane reduce)
      C trp = cmk(0.f, 0.f);
      for (int e = lane; e < 64; e += 32) {
        int i = e >> 3, j = e & 7;
        trp = cadd(trp, cmul(cmk(sMre[i*8+j], sMim[i*8+j]), cmk(sVre[j*8+i], sVim[j*8+i])));
      }
      C tr = cmk(wsum32(trp.re), wsum32(trp.im));

      // uV[m][q], uVmk[m][q]
      {
        int m = lane >> 3, q = lane & 7;
        C acc = cmk(0.f, 0.f);
        #pragma unroll
        for (int r = 0; r < 8; ++r)
          acc = cadd(acc, cmul(cmk(sUre[m*8+r], sUim[m*8+r]), cmk(sVre[r*8+q], sVim[r*8+q])));
        sUVre[lane] = acc.re; sUVim[lane] = acc.im;
        C uw = cmk(0.f, 0.f);
        #pragma unroll
        for (int r = 0; r < 8; ++r)
          uw = cadd(uw, cmul(cmk(sUre[m*8+r], sUim[m*8+r]), cmk(sWre[r*4+m], sWim[r*4+m])));
        C t2 = cmulc(uw, cmk(sWre[q*4+m], sWim[q*4+m]));
        sUKre[lane] = acc.re - t2.re; sUKim[lane] = acc.im - t2.im;
      }
      __syncthreads();

      if (lane < 4) {
        C q1 = cmk(0.f, 0.f), q2 = cmk(0.f, 0.f);
        #pragma unroll
        for (int q = 0; q < 8; ++q) {
          C u = cmk(sUre[lane*8+q], sUim[lane*8+q]);
          q1 = cadd(q1, cmulc(cmk(sUVre[lane*8+q], sUVim[lane*8+q]), u));
          q2 = cadd(q2, cmulc(cmk(sUKre[lane*8+q], sUKim[lane*8+q]), u));
        }
        C d1 = cmk(fmaf(K_LN2, q1.re + K_SIGMA2, 1e-4f), K_LN2*q1.im);
        C d2 = cmk(fmaf(K_LN2, q2.re + K_SIGMA2, 1e-4f), K_LN2*q2.im);
        C i1 = cdiv(cmk(1.f, 0.f), d1);
        C i2 = cdiv(cmk(1.f, 0.f), d2);
        sI1re[lane]=i1.re; sI1im[lane]=i1.im; sI2re[lane]=i2.re; sI2im[lane]=i2.im;
      }
      __syncthreads();

      // F update: lane owns rows n = 2*lane, 2*lane+1
      const float stepF = ss[jj*20 + ii*2 + 0];
      for (int th = 0; th < 2; ++th) {
        int n = 2*lane + th;
        #pragma unroll
        for (int r = 0; r < 8; ++r) {
          C gc = cmk(0.f, 0.f);
          #pragma unroll
          for (int m = 0; m < 4; ++m) {
            C h  = cmk(sHre[m*64+n], sHim[m*64+n]);
            C a1 = cmul(cmk(sI1re[m], sI1im[m]), cmul(h, cmk(sUVre[m*8+r], sUVim[m*8+r])));
            C a2 = cmul(cmk(sI2re[m], sI2im[m]), cmul(h, cmk(sUKre[m*8+r], sUKim[m*8+r])));
            gc = cadd(gc, csub(a1, a2));
          }
          C num = cmk(0.f, 0.f);
          #pragma unroll
          for (int q = 0; q < 8; ++q) {
            C mf = cmk(sP[n*16+q] - sQ[n*16+q+8], sP[n*16+q+8] + sQ[n*16+q]);
            num = cadd(num, cmul(mf, cmk(sVre[q*8+r], sVim[q*8+r])));
          }
          C g = cadd(gc, cscale(cdiv(num, tr), 0.01f));
          FRE(n,r) += stepF * g.re;
          FIM(n,r) += stepF * g.im;
        }
      }
      __syncthreads();

      // power scale F: p = sum |F W|^2
      float pp = 0.f;
      for (int n = lane; n < 64; n += 32) {
        #pragma unroll
        for (int m = 0; m < 4; ++m) {
          C fw = cmk(0.f, 0.f);
          #pragma unroll
          for (int r = 0; r < 8; ++r)
            fw = cadd(fw, cmul(cmk(FRE(n,r), FIM(n,r)), cmk(sWre[r*4+m], sWim[r*4+m])));
          pp = fmaf(fw.re, fw.re, fmaf(fw.im, fw.im, pp));
        }
      }
      float scal = sqrtf(Pt / (wsum32(pp) + 1e-12f));
      for (int e = lane*32; e < lane*32 + 32; ++e) sFC[e] *= scal;   // own rows 2l,2l+1
      __syncthreads();
    }

    // ---------------- projection + W update ----------------
    for (int th = 0; th < 2; ++th) {
      int n = 2*lane + th;
      #pragma unroll
      for (int r = 0; r < 8; ++r) {
        float fre = FRE(n,r), fim = FIM(n,r);
        float mag = sqrtf(fre*fre + fim*fim);
        if (mag > 1e-12f) { FRE(n,r) = fre/mag; FIM(n,r) = fim/mag; }
        else              { FRE(n,r) = 0.f;     FIM(n,r) = 0.f; }
      }
    }
    __syncthreads();

    // g[m][r] = sum_n conj(F[n][r]) H[m][n]
    {
      int m = lane >> 3, r = lane & 7;
      C acc = cmk(0.f, 0.f);
      for (int n = 0; n < 64; ++n)
        acc = cadd(acc, cmulc(cmk(sHre[m*64+n], sHim[m*64+n]), cmk(FRE(n,r), FIM(n,r))));
      sGre[lane] = acc.re; sGim[lane] = acc.im;
    }
    __syncthreads();
    if (lane < 16) {   // gHW[m][k] = sum_r conj(g[m][r]) W[r][k]
      int m = lane >> 2, k = lane & 3;
      C acc = cmk(0.f, 0.f);
      #pragma unroll
      for (int r = 0; r < 8; ++r)
        acc = cadd(acc, cmulc(cmk(sWre[r*4+k], sWim[r*4+k]), cmk(sGre[m*8+r], sGim[m*8+r])));
      sGHre[lane] = acc.re; sGHim[lane] = acc.im;
    }
    __syncthreads();
    if (lane < 4) {
      float s = 0.f;
      #pragma unroll
      for (int k = 0; k < 4; ++k)
        s = fmaf(sGHre[lane*4+k], sGHre[lane*4+k], fmaf(sGHim[lane*4+k], sGHim[lane*4+k], s));
      float denom = K_LN2 * (s + K_SIGMA2);
      sFcRe[lane] = sGHre[lane*4+lane] / denom;
      sFcIm[lane] = sGHim[lane*4+lane] / denom;
    }
    __syncthreads();

    // FHMF with projected F (WMMA again), V with current W
    compute_MmF(sMmRe, sMmIm, sFC, sP, sQ, lane);
    for (int e = lane; e < 64; e += 32) {
      int i = e >> 3, j = e & 7;
      C acc = cmk(0.f, 0.f);
      #pragma unroll
      for (int m = 0; m < 4; ++m)
        acc = cadd(acc, cmulc(cmk(sWre[i*4+m], sWim[i*4+m]), cmk(sWre[j*4+m], sWim[j*4+m])));
      sVre[e] = acc.re; sVim[e] = acc.im;
    }
    for (int e = lane; e < 64; e += 32) {
      int i = e >> 3, j = e & 7;
      C acc = cmk(0.f, 0.f);
      for (int n = 0; n < 64; ++n) {
        C mf = cmk(sP[n*16+j] - sQ[n*16+j+8], sP[n*16+j+8] + sQ[n*16+j]);
        acc = cadd(acc, cmulc(mf, cmk(FRE(n,i), FIM(n,i))));
      }
      sMre[e] = acc.re; sMim[e] = acc.im;
    }
    __syncthreads();
    C trp = cmk(0.f, 0.f);
    for (int e = lane; e < 64; e += 32) {
      int i = e >> 3, j = e & 7;
      trp = cadd(trp, cmul(cmk(sMre[i*8+j], sMim[i*8+j]), cmk(sVre[j*8+i], sVim[j*8+i])));
    }
    C tr2 = cmk(wsum32(trp.re), wsum32(trp.im));

    // W update: lane owns entry (r = lane/4, m = lane%4)
    const float stepW = ss[ii*2 + 1];
    C gw;
    {
      int r = lane >> 2, m = lane & 3;
      C gwc = cscale(cmul(cmk(sGre[m*8+r], sGim[m*8+r]), cmk(sFcRe[m], sFcIm[m])), 0.25f);
      C fw = cmk(0.f, 0.f);
      #pragma unroll
      for (int j = 0; j < 8; ++j)
        fw = cadd(fw, cmul(cmk(sMre[r*8+j], sMim[r*8+j]), cmk(sWre[j*4+m], sWim[j*4+m])));
      C gwr = cdiv(fw, tr2);
      gw = cadd(gwc, cscale(gwr, 0.01f));
    }
    __syncthreads();
    sWre[lane] += stepW * gw.re;
    sWim[lane] += stepW * gw.im;
    __syncthreads();

    // power scale W (with new W)
    float pp = 0.f;
    for (int n = lane; n < 64; n += 32) {
      #pragma unroll
      for (int m = 0; m < 4; ++m) {
        C fw = cmk(0.f, 0.f);
        #pragma unroll
        for (int r = 0; r < 8; ++r)
          fw = cadd(fw, cmul(cmk(FRE(n,r), FIM(n,r)), cmk(sWre[r*4+m], sWim[r*4+m])));
        pp = fmaf(fw.re, fw.re, fmaf(fw.im, fw.im, pp));
      }
    }
    float scal = sqrtf(Pt / (wsum32(pp) + 1e-12f));
    sWre[lane] *= scal; sWim[lane] *= scal;
    __syncthreads();
  }

  // output: [F.re(512) | F.im(512) | W.re(32) | W.im(32)] per batch
  float* ob = outg + (size_t)b * 1088;
  for (int e = lane; e < 512; e += 32) {
    int n = e >> 3, r = e & 7;
    ob[e]       = FRE(n,r);
    ob[512 + e] = FIM(n,r);
  }
  ob[1024 + lane] = sWre[lane];
  ob[1056 + lane] = sWim[lane];
}

extern "C" void kernel_launch(void* const* d_in, const int* in_sizes, int n_in,
                              void* d_out, int out_size, void* d_ws, size_t ws_size,
                              hipStream_t stream) {
  const float* H  = (const float*)d_in[0];   // [1,1024,4,64] complex64
  const float* Ad = (const float*)d_in[1];   // [64,64] complex64
  const float* Rn = (const float*)d_in[2];   // [64,64] complex64
  const float* ss = (const float*)d_in[3];   // [5,10,2] float32
  const float* F0 = (const float*)d_in[4];   // [1,1024,64,8] complex64
  const float* W0 = (const float*)d_in[5];   // [1,1024,8,4] complex64
  const float* Pt = (const float*)d_in[6];   // scalar
  float* out = (float*)d_out;

  float* Tws = (float*)d_ws;            // 64*64*2 floats
  float* Mm  = Tws + 64*64*2;           // 64*64*2 floats

  const int B = out_size / 1088;        // S*B problems

  k_mm_RA <<<(4096 + 127) / 128, 128, 0, stream>>>(Rn, Ad, Tws);
  k_mm_AHT<<<(4096 + 127) / 128, 128, 0, stream>>>(Ad, Tws, Mm);
  k_pga   <<<B, 32, 0, stream>>>(H, Mm, ss, F0, W0, Pt, out, B);
}